// RGCN_20117626814888
// MI455X (gfx1250) — compile-verified
//
#include <hip/hip_runtime.h>

#define N_NODES 50000
#define N_EDGES 800000
#define N_REL   65
#define DIM     64
// worst-case padded edge count: E + 15 padding per relation, rounded to tiles
#define MAX_TILES 50061
#define PADMAX (MAX_TILES * 16)

typedef float v2f __attribute__((ext_vector_type(2)));
typedef float v8f __attribute__((ext_vector_type(8)));

// ---------------- setup kernels ----------------

__global__ void zero_f32v4_kernel(float4* __restrict__ p, int n4) {
    int i = blockIdx.x * blockDim.x + threadIdx.x;
    if (i < n4) p[i] = make_float4(0.f, 0.f, 0.f, 0.f);
}

__global__ void count_hist_kernel(const int* __restrict__ dst,
                                  const int* __restrict__ et,
                                  float* __restrict__ cnt,
                                  int* __restrict__ hist) {
    int i = blockIdx.x * blockDim.x + threadIdx.x;
    if (i >= N_EDGES) return;
    int t = et[i];
    atomicAdd(&cnt[dst[i] * N_REL + t], 1.0f);
    atomicAdd(&hist[t], 1);
}

// exclusive prefix over 65 bins, each rounded up to a multiple of 16
__global__ void prefix_kernel(const int* __restrict__ hist,
                              int* __restrict__ off,
                              int* __restrict__ woff) {
    if (threadIdx.x == 0 && blockIdx.x == 0) {
        int run = 0;
        for (int r = 0; r < N_REL; ++r) {
            off[r]  = run;
            woff[r] = run;
            run += ((hist[r] + 15) >> 4) << 4;
        }
        off[N_REL] = run;  // total padded edges (multiple of 16)
    }
}

__global__ void tile_rel_kernel(const int* __restrict__ off,
                                int* __restrict__ tile_rel) {
    int t = blockIdx.x * blockDim.x + threadIdx.x;
    if (t >= MAX_TILES) return;
    int e = t << 4;
    if (e >= off[N_REL]) { tile_rel[t] = -1; return; }
    int lo = 0, hi = N_REL - 1;           // largest r with off[r] <= e
    while (lo < hi) {
        int mid = (lo + hi + 1) >> 1;
        if (off[mid] <= e) lo = mid; else hi = mid - 1;
    }
    tile_rel[t] = lo;
}

__global__ void scatter_kernel(const int* __restrict__ src,
                               const int* __restrict__ dst,
                               const int* __restrict__ et,
                               const float* __restrict__ cnt,
                               int* __restrict__ woff,
                               int* __restrict__ ssrc,
                               int* __restrict__ sdst,
                               float* __restrict__ sscale) {
    int i = blockIdx.x * blockDim.x + threadIdx.x;
    if (i >= N_EDGES) return;
    int t = et[i];
    int d = dst[i];
    int pos = atomicAdd(&woff[t], 1);
    ssrc[pos]   = src[i];
    sdst[pos]   = d;
    sscale[pos] = 1.0f / fmaxf(cnt[d * N_REL + t], 1.0f);
}

// ---------------- WMMA compute kernels ----------------

// One wave per 16-edge tile: D(16x64) = A(16 edges x 64) * W[r](64x64),
// per-edge mean scale folded into the output, atomic-accumulated into conv.
__global__ void __launch_bounds__(32)
edge_wmma_kernel(const float* __restrict__ x,
                 const int* __restrict__ ssrc,
                 const int* __restrict__ sdst,
                 const float* __restrict__ sscale,
                 const int* __restrict__ tile_rel,
                 const float* __restrict__ W,
                 float* __restrict__ conv) {
    int tile = blockIdx.x;
    int r = tile_rel[tile];
    if (r < 0) return;                       // wave-uniform exit; EXEC stays full
    int lane  = threadIdx.x & 31;
    int laneM = lane & 15;
    int khalf = lane >> 4;                   // 0: K%4 in {0,1}; 1: K%4 in {2,3}
    int ebase = tile << 4;

    const float* xrow = x + (size_t)ssrc[ebase + laneM] * DIM;
    const float* Wr   = W + (size_t)r * DIM * DIM + laneM;

    v8f c[4] = {};
#pragma unroll
    for (int k = 0; k < 16; ++k) {
        int kb = 4 * k;
        v2f a = *(const v2f*)(xrow + kb + 2 * khalf);
        const float* w0 = Wr + (size_t)(kb + khalf) * DIM;      // B row K=kb+khalf
        const float* w1 = Wr + (size_t)(kb + 2 + khalf) * DIM;  // B row K=kb+2+khalf
#pragma unroll
        for (int nt = 0; nt < 4; ++nt) {
            v2f b;
            b.x = w0[nt * 16];
            b.y = w1[nt * 16];
            c[nt] = __builtin_amdgcn_wmma_f32_16x16x4_f32(false, a, false, b,
                                                          (short)0, c[nt],
                                                          false, false);
        }
    }
#pragma unroll
    for (int i = 0; i < 8; ++i) {
        int m = ebase + i + (khalf << 3);    // C layout: vgpr i = rows i / i+8
        int drow = sdst[m];
        float sc = sscale[m];                // 0 for padding rows -> adds exact 0
        float* cr = conv + (size_t)drow * DIM + laneM;
#pragma unroll
        for (int nt = 0; nt < 4; ++nt)
            atomicAdd(cr + nt * 16, sc * c[nt][i]);
    }
}

// out = relu(conv + xin@root + b) + relu(xin@res + rb)
// One wave per 16-row tile, full 64 cols, both GEMMs sharing A fragments.
__global__ void __launch_bounds__(32)
combine_wmma_kernel(const float* __restrict__ xin,
                    const float* __restrict__ conv,
                    const float* __restrict__ rootW,
                    const float* __restrict__ resW,
                    const float* __restrict__ bias,
                    const float* __restrict__ rbias,
                    float* __restrict__ out) {
    int m0    = blockIdx.x << 4;
    int lane  = threadIdx.x & 31;
    int laneM = lane & 15;
    int khalf = lane >> 4;

    const float* xrow = xin + (size_t)(m0 + laneM) * DIM;
    const float* R0 = rootW + laneM;
    const float* R1 = resW  + laneM;

    v8f c0[4] = {};   // root-GEMM accumulators, n-tiles 0..3
    v8f c1[4] = {};   // residual-GEMM accumulators
#pragma unroll
    for (int k = 0; k < 16; ++k) {
        int kb = 4 * k;
        v2f a = *(const v2f*)(xrow + kb + 2 * khalf);
        const float* r0a = R0 + (size_t)(kb + khalf) * DIM;
        const float* r0b = R0 + (size_t)(kb + 2 + khalf) * DIM;
        const float* r1a = R1 + (size_t)(kb + khalf) * DIM;
        const float* r1b = R1 + (size_t)(kb + 2 + khalf) * DIM;
#pragma unroll
        for (int nt = 0; nt < 4; ++nt) {
            v2f b0, b1;
            b0.x = r0a[nt * 16]; b0.y = r0b[nt * 16];
            b1.x = r1a[nt * 16]; b1.y = r1b[nt * 16];
            c0[nt] = __builtin_amdgcn_wmma_f32_16x16x4_f32(false, a, false, b0,
                                                           (short)0, c0[nt],
                                                           false, false);
            c1[nt] = __builtin_amdgcn_wmma_f32_16x16x4_f32(false, a, false, b1,
                                                           (short)0, c1[nt],
                                                           false, false);
        }
    }
    float bn[4], rbn[4];
#pragma unroll
    for (int nt = 0; nt < 4; ++nt) {
        bn[nt]  = bias[nt * 16 + laneM];
        rbn[nt] = rbias[nt * 16 + laneM];
    }
#pragma unroll
    for (int i = 0; i < 8; ++i) {
        int m = m0 + i + (khalf << 3);
        const float* convr = conv + (size_t)m * DIM + laneM;
        float*       outr  = out  + (size_t)m * DIM + laneM;
#pragma unroll
        for (int nt = 0; nt < 4; ++nt) {
            float v0 = c0[nt][i] + convr[nt * 16] + bn[nt];
            float v1 = c1[nt][i] + rbn[nt];
            outr[nt * 16] = fmaxf(v0, 0.0f) + fmaxf(v1, 0.0f);
        }
    }
}

// ---------------- host launcher ----------------

extern "C" void kernel_launch(void* const* d_in, const int* in_sizes, int n_in,
                              void* d_out, int out_size, void* d_ws, size_t ws_size,
                              hipStream_t stream) {
    (void)in_sizes; (void)n_in; (void)out_size; (void)ws_size;

    const float* x     = (const float*)d_in[0];
    const int*   src   = (const int*)d_in[1];           // edge_index row 0
    const int*   dst   = (const int*)d_in[1] + N_EDGES; // edge_index row 1
    const int*   etype = (const int*)d_in[2];
    const float* W1    = (const float*)d_in[3];
    const float* root1 = (const float*)d_in[4];
    const float* b1    = (const float*)d_in[5];
    const float* rw1   = (const float*)d_in[6];
    const float* rb1   = (const float*)d_in[7];
    const float* W2    = (const float*)d_in[8];
    const float* root2 = (const float*)d_in[9];
    const float* b2    = (const float*)d_in[10];
    const float* rw2   = (const float*)d_in[11];
    const float* rb2   = (const float*)d_in[12];
    float* out = (float*)d_out;

    // workspace layout (4-byte words)
    float* cnt      = (float*)d_ws;                          // N*R
    int*   hist     = (int*)(cnt + (size_t)N_NODES * N_REL); // 65
    int*   off      = hist + N_REL;                          // 66
    int*   woff     = off + N_REL + 1;                       // 65
    int*   tile_rel = woff + N_REL;                          // MAX_TILES
    int*   ssrc     = tile_rel + MAX_TILES;                  // PADMAX
    int*   sdst     = ssrc + PADMAX;                         // PADMAX
    float* sscale   = (float*)(sdst + PADMAX);               // PADMAX
    float* conv     = sscale + PADMAX;                       // N*DIM
    float* h1       = conv + (size_t)N_NODES * DIM;          // N*DIM

    const int ZB = 256;
    // all zeroed regions are multiples of 4 floats
    {
        int n4 = (N_NODES * N_REL) / 4;
        zero_f32v4_kernel<<<(n4 + ZB - 1) / ZB, ZB, 0, stream>>>((float4*)cnt, n4);
    }
    zero_f32v4_kernel<<<1, ZB, 0, stream>>>((float4*)hist, 196 / 4);  // hist+off+woff
    {
        int n4 = (3 * PADMAX) / 4;
        zero_f32v4_kernel<<<(n4 + ZB - 1) / ZB, ZB, 0, stream>>>((float4*)ssrc, n4);
    }

    count_hist_kernel<<<(N_EDGES + ZB - 1) / ZB, ZB, 0, stream>>>(dst, etype, cnt, hist);
    prefix_kernel<<<1, 1, 0, stream>>>(hist, off, woff);
    tile_rel_kernel<<<(MAX_TILES + ZB - 1) / ZB, ZB, 0, stream>>>(off, tile_rel);
    scatter_kernel<<<(N_EDGES + ZB - 1) / ZB, ZB, 0, stream>>>(src, dst, etype, cnt, woff,
                                                               ssrc, sdst, sscale);

    int conv4 = (N_NODES * DIM) / 4;
    dim3 egrid(MAX_TILES);
    dim3 cgrid(N_NODES / 16);

    // ---- layer 1 ----
    zero_f32v4_kernel<<<(conv4 + ZB - 1) / ZB, ZB, 0, stream>>>((float4*)conv, conv4);
    edge_wmma_kernel<<<egrid, 32, 0, stream>>>(x, ssrc, sdst, sscale, tile_rel, W1, conv);
    combine_wmma_kernel<<<cgrid, 32, 0, stream>>>(x, conv, root1, rw1, b1, rb1, h1);

    // ---- layer 2 ----
    zero_f32v4_kernel<<<(conv4 + ZB - 1) / ZB, ZB, 0, stream>>>((float4*)conv, conv4);
    edge_wmma_kernel<<<egrid, 32, 0, stream>>>(h1, ssrc, sdst, sscale, tile_rel, W2, conv);
    combine_wmma_kernel<<<cgrid, 32, 0, stream>>>(h1, conv, root2, rw2, b2, rb2, out);
}